// ProtoNet_83726092469058
// MI455X (gfx1250) — compile-verified
//
#include <hip/hip_runtime.h>
#include <math.h>

// ---------------- problem constants ----------------
#define SHOTN 5
#define WAYN  32
#define DIMN  640
#define NQ    4096
#define RN    10
#define TEMPF 16.0f
#define EPSF  1e-05f
#define NALL  (SHOTN*WAYN + NQ)   // 4256

// ---------------- workspace layout (float offsets) ----------------
#define OFF_P    0                       // protos            32*640
#define OFF_A    20480                   // all_data          640
#define OFF_QN2  21120                   // |q|^2             4096
#define OFF_C    25216                   // c                 32
#define OFF_SC   25248                   // sqrt(c)           32
#define OFF_MAXN 25280                   // clip bound        32
#define OFF_U2P  25312                   // |proto_p|^2       32
#define OFF_PP   25344                   // scaled protos     32*640
#define OFF_DIS  45824                   // dis_mat           32*4096
#define OFF_S    176896                  // row sums          32
#define OFF_T    176928                  // first-10-col sums 32
#define OFF_NID  176960                  // n_i_d             32*10
#define OFF_NEAR 177280                  // near idx (int)    32*10
#define OFF_TST  177600                  // scaled test proto 32*640
#define OFF_TN2  198080                  // |test_p|^2        32

typedef __attribute__((ext_vector_type(2))) float v2f;
typedef __attribute__((ext_vector_type(8))) float v8f;

// ---------------- K0: protos + all_data (column means) ----------------
__global__ void k_prep(const float* __restrict__ shot,
                       const float* __restrict__ query,
                       float* __restrict__ ws) {
    int d = blockIdx.x * blockDim.x + threadIdx.x;
    if (d >= DIMN) return;
    float tot = 0.f;
    for (int w = 0; w < WAYN; ++w) {
        float s = 0.f;
        for (int sh = 0; sh < SHOTN; ++sh)
            s += shot[(sh * WAYN + w) * DIMN + d];
        ws[OFF_P + w * DIMN + d] = s * (1.0f / SHOTN);
        tot += s;
    }
    for (int q = 0; q < NQ; ++q)
        tot += query[q * DIMN + d];
    ws[OFF_A + d] = tot * (1.0f / NALL);
}

// ---------------- K0b: query squared norms ----------------
__global__ void k_qn2(const float* __restrict__ query, float* __restrict__ ws) {
    int wave = threadIdx.x >> 5, lane = threadIdx.x & 31;
    int r = blockIdx.x * (blockDim.x >> 5) + wave;
    if (r >= NQ) return;
    const float* row = query + (size_t)r * DIMN;
    float s = 0.f;
    for (int d = lane; d < DIMN; d += 32) { float v = row[d]; s += v * v; }
    for (int off = 16; off; off >>= 1) s += __shfl_xor(s, off, 32);
    if (lane == 0) ws[OFF_QN2 + r] = s;
}

// expmap0 scale factor: y = alpha * x  (alpha depends on ||x|| and c only)
__device__ __forceinline__ float expmap_scale(float nx, float sc, float maxn) {
    float n    = fmaxf(nx, EPSF);
    float t    = tanhf(sc * n);
    float base = t / (sc * n);
    float ny   = fmaxf(base * nx, EPSF);
    return base * (ny > maxn ? maxn / ny : 1.0f);
}

// Mobius distance from scalars: dot = x.y, u2 = |x|^2, y2 = |y|^2
__device__ __forceinline__ float mob_dist(float dot, float u2, float y2,
                                          float cc, float sc) {
    float aa   = 1.f - 2.f * cc * dot + cc * y2;
    float bb   = 1.f - cc * u2;
    float num2 = aa * aa * u2 - 2.f * aa * bb * dot + bb * bb * y2;
    float den  = fmaxf(1.f - 2.f * cc * dot + cc * cc * u2 * y2, EPSF);
    float n    = sqrtf(fmaxf(num2, 0.f)) / den;
    float arg  = fminf(fmaxf(sc * n, 0.f), 1.f - EPSF);
    // atanh(arg) = 0.5*ln((1+arg)/(1-arg))
    return (2.f / sc) * 0.5f * logf((1.f + arg) / (1.f - arg));
}

// ---------------- K1: controller MLP -> c, scales, scaled protos ----------------
__global__ void k_ctrl(const float* __restrict__ W1, const float* __restrict__ b1,
                       const float* __restrict__ W2, const float* __restrict__ b2,
                       const float* __restrict__ W3, const float* __restrict__ b3,
                       float* __restrict__ ws) {
    __shared__ float h1[WAYN * 128];
    __shared__ float h2[WAYN * 64];
    __shared__ float sal[WAYN];
    const float* P = ws + OFF_P;
    const float* A = ws + OFF_A;
    int tid = threadIdx.x;

    for (int o = tid; o < WAYN * 128; o += 256) {
        int w = o >> 7, j = o & 127;
        float acc = b1[j];
        for (int k = 0; k < DIMN; ++k) acc += P[w * DIMN + k] * W1[k * 128 + j];
        for (int k = 0; k < DIMN; ++k) acc += A[k] * W1[(DIMN + k) * 128 + j];
        h1[o] = fmaxf(acc, 0.f);
    }
    __syncthreads();
    for (int o = tid; o < WAYN * 64; o += 256) {
        int w = o >> 6, j = o & 63;
        float acc = b2[j];
        for (int k = 0; k < 128; ++k) acc += h1[w * 128 + k] * W2[k * 64 + j];
        h2[o] = fmaxf(acc, 0.f);
    }
    __syncthreads();
    if (tid < WAYN) {
        int w = tid;
        float o3[5]; float mx = -1e30f;
        for (int j = 0; j < 5; ++j) {
            float acc = b3[j];
            for (int k = 0; k < 64; ++k) acc += h2[w * 64 + k] * W3[k * 5 + j];
            o3[j] = acc; mx = fmaxf(mx, acc);
        }
        float se = 0.f;
        for (int j = 0; j < 5; ++j) { o3[j] = expf(o3[j] - mx); se += o3[j]; }
        float c = 0.f;
        for (int j = 0; j < 5; ++j) c += (o3[j] / se) * ((j + 1) * 0.2f); // L/DIV=0.2
        float sc   = sqrtf(c);
        float maxn = (1.0f - 0.001f) / sc;
        float pn2 = 0.f;
        for (int d = 0; d < DIMN; ++d) { float v = P[w * DIMN + d]; pn2 += v * v; }
        float nx = sqrtf(pn2);
        float al = expmap_scale(nx, sc, maxn);
        ws[OFF_C + w] = c; ws[OFF_SC + w] = sc; ws[OFF_MAXN + w] = maxn;
        float pr = al * nx;
        ws[OFF_U2P + w] = pr * pr;
        sal[w] = al;
    }
    __syncthreads();
    float* Pp = ws + OFF_PP;
    for (int i = tid; i < WAYN * DIMN; i += 256) Pp[i] = P[i] * sal[i / DIMN];
}

// ---------------- shared WMMA GEMM core (M=32 x N=16/wave, K=640) ----------------
// A rows = 32 pre-scaled prototype-like vectors, B cols = raw queries.
// f32 WMMA 16x16x4: A lane(l): m=l%16, kh=l/16 -> A[m][k+2kh..+1]
//                   B lane(l): n=l%16, kh=l/16 -> B[k+2kh..+1][n]
//                   C VGPR r, lane l: M=r+8*(l/16), N=l%16
__global__ __launch_bounds__(256)
void k_gemm_dis(const float* __restrict__ query, float* __restrict__ ws) {
    const float* Pp  = ws + OFF_PP;
    float*       dis = ws + OFF_DIS;
    int wave = threadIdx.x >> 5, lane = threadIdx.x & 31;
    int col0 = blockIdx.x * 128 + wave * 16;
    int nn = lane & 15, kh = lane >> 4;
    const float* brow = query + (size_t)(col0 + nn) * DIMN + 2 * kh;
    int   q   = col0 + nn;
    float qn2 = ws[OFF_QN2 + q];
    float nx  = sqrtf(qn2);

    for (int mt = 0; mt < 2; ++mt) {
        const float* arow = Pp + (size_t)(mt * 16 + nn) * DIMN + 2 * kh;
        v8f acc = {};
#pragma unroll 4
        for (int k = 0; k < DIMN; k += 4) {
            v2f a = *(const v2f*)(arow + k);
            v2f b = *(const v2f*)(brow + k);
            __builtin_prefetch(brow + k + 64, 0, 0);
            acc = __builtin_amdgcn_wmma_f32_16x16x4_f32(
                false, a, false, b, (short)0, acc, false, false);
        }
        for (int r = 0; r < 8; ++r) {
            int   w    = mt * 16 + (kh << 3) + r;
            float sc   = ws[OFF_SC + w];
            float cc   = ws[OFF_C + w];
            float maxn = ws[OFF_MAXN + w];
            float u2   = ws[OFF_U2P + w];
            float beta = expmap_scale(nx, sc, maxn);
            float y2   = beta * beta * qn2;
            float dot  = beta * acc[r];
            dis[w * NQ + q] = mob_dist(dot, u2, y2, cc, sc);
        }
    }
}

// ---------------- K3: per-way row sum, T, stable top-10 ----------------
__global__ void k_rowstats(float* __restrict__ ws) {
    int w = blockIdx.x;
    const float* row = ws + OFF_DIS + (size_t)w * NQ;
    __shared__ float sv[256];
    __shared__ int   si[256];
    __shared__ float redf[256];
    __shared__ int   chosen[RN];
    int tid = threadIdx.x;

    float s = 0.f;
    for (int q = tid; q < NQ; q += 256) s += row[q];
    redf[tid] = s; __syncthreads();
    for (int off = 128; off; off >>= 1) {
        if (tid < off) redf[tid] += redf[tid + off];
        __syncthreads();
    }
    if (tid == 0) {
        ws[OFF_S + w] = redf[0];
        float t = 0.f;
        for (int q = 0; q < RN; ++q) t += row[q];
        ws[OFF_T + w] = t;
    }
    __syncthreads();

    for (int r = 0; r < RN; ++r) {
        float mv = 3.4e38f; int mi = 0x7fffffff;
        for (int q = tid; q < NQ; q += 256) {
            bool used = false;
            for (int j = 0; j < r; ++j) used |= (chosen[j] == q);
            if (!used) {
                float v = row[q];
                if (v < mv || (v == mv && q < mi)) { mv = v; mi = q; }
            }
        }
        sv[tid] = mv; si[tid] = mi; __syncthreads();
        for (int off = 128; off; off >>= 1) {
            if (tid < off) {
                float v2 = sv[tid + off]; int i2 = si[tid + off];
                if (v2 < sv[tid] || (v2 == sv[tid] && i2 < si[tid])) {
                    sv[tid] = v2; si[tid] = i2;
                }
            }
            __syncthreads();
        }
        if (tid == 0) {
            chosen[r] = si[0];
            ((int*)(ws + OFF_NEAR))[w * RN + r] = si[0];
            ws[OFF_NID + w * RN + r] = sv[0];
        }
        __syncthreads();
    }
}

// ---------------- K4: relation net + new (scaled) test prototypes ----------------
__global__ void k_relnet(const float* __restrict__ query,
                         const float* __restrict__ Wr1, const float* __restrict__ br1,
                         const float* __restrict__ Wr2, const float* __restrict__ br2,
                         float* __restrict__ ws) {
    __shared__ float rrin[WAYN * 22];
    __shared__ float pre[WAYN * RN];
    __shared__ float hr[WAYN * RN];
    __shared__ float iw[WAYN * RN];
    __shared__ float onw[WAYN];
    __shared__ float a2[WAYN];
    int tid = threadIdx.x;
    const int*   nearI = (const int*)(ws + OFF_NEAR);
    const float* nid   = ws + OFF_NID;
    const float* dis   = ws + OFF_DIS;

    for (int o = tid; o < WAYN * RN; o += 256) {
        int w = o / RN, k = o % RN;
        int qi = nearI[o];
        float full = 0.f, prefix = 0.f;
        for (int j = 0; j < WAYN; ++j) {
            float v = dis[(size_t)j * NQ + qi];
            full += v; if (j < w) prefix += v;
        }
        rrin[w * 22 + k]      = nid[o];
        rrin[w * 22 + RN + k] = (full - nid[o]) / (WAYN - 1);
        pre[o] = prefix;
        (void)k;
    }
    __syncthreads();
    if (tid < WAYN) {
        int w = tid;
        float sn = 0.f;
        for (int k = 0; k < RN; ++k) sn += nid[w * RN + k];
        float S = ws[OFF_S + w];
        rrin[w * 22 + 2 * RN] = (S - sn) / (float)(NQ - RN);   // o_i_d
        float sumA = 0.f;
        for (int j = 0; j < w; ++j) sumA += ws[OFF_S + j];
        float psum = 0.f;
        for (int k = 0; k < RN; ++k) psum += pre[w * RN + k];
        sumA -= psum;
        float sumB = 0.f;
        for (int j = w + 1; j < WAYN; ++j) sumB += ws[OFF_S + j] - ws[OFF_T + j];
        rrin[w * 22 + 2 * RN + 1] = (sumA + sumB) / ((WAYN - 1.f) * (NQ - RN));
    }
    __syncthreads();
    for (int o = tid; o < WAYN * RN; o += 256) {
        int w = o / RN, j = o % RN;
        float acc = br1[j];
        for (int k = 0; k < 22; ++k) acc += rrin[w * 22 + k] * Wr1[k * RN + j];
        hr[o] = fmaxf(acc, 0.f);
    }
    __syncthreads();
    if (tid < WAYN) {
        int w = tid;
        float orr[RN + 1]; float mx = -1e30f;
        for (int j = 0; j < RN + 1; ++j) {
            float acc = br2[j];
            for (int k = 0; k < RN; ++k) acc += hr[w * RN + k] * Wr2[k * (RN + 1) + j];
            orr[j] = acc;
            if (j < RN) mx = fmaxf(mx, acc);
        }
        float se = 0.f;
        for (int j = 0; j < RN; ++j) { orr[j] = expf(orr[j] - mx); se += orr[j]; }
        for (int j = 0; j < RN; ++j) iw[w * RN + j] = orr[j] / se;
        onw[w] = 1.f / (1.f + expf(-orr[RN]));
    }
    __syncthreads();
    float* tst = ws + OFF_TST;
    const float* P = ws + OFF_P;
    for (int i = tid; i < WAYN * DIMN; i += 256) {
        int w = i / DIMN, d = i % DIMN;
        float wd = 0.f;
        for (int k = 0; k < RN; ++k)
            wd += iw[w * RN + k] * query[(size_t)nearI[w * RN + k] * DIMN + d];
        float ov = onw[w];
        tst[i] = P[i] * ov + wd * (1.f - ov);
    }
    __syncthreads();
    if (tid < WAYN) {
        int w = tid;
        float n2 = 0.f;
        for (int d = 0; d < DIMN; ++d) { float v = tst[w * DIMN + d]; n2 += v * v; }
        float nx = sqrtf(n2);
        float sc = ws[OFF_SC + w];
        float al = expmap_scale(nx, sc, ws[OFF_MAXN + w]);
        a2[w] = al;
        float tp = al * nx;
        ws[OFF_TN2 + w] = tp * tp;
    }
    __syncthreads();
    for (int i = tid; i < WAYN * DIMN; i += 256) tst[i] *= a2[i / DIMN];
}

// ---------------- K5: second WMMA GEMM + output (-dist.T/TEMP) ----------------
__global__ __launch_bounds__(256)
void k_gemm_out(const float* __restrict__ query, const float* __restrict__ ws,
                float* __restrict__ out) {
    const float* Tp = ws + OFF_TST;
    int wave = threadIdx.x >> 5, lane = threadIdx.x & 31;
    int col0 = blockIdx.x * 128 + wave * 16;
    int nn = lane & 15, kh = lane >> 4;
    const float* brow = query + (size_t)(col0 + nn) * DIMN + 2 * kh;
    int   q   = col0 + nn;
    float qn2 = ws[OFF_QN2 + q];
    float nx  = sqrtf(qn2);

    for (int mt = 0; mt < 2; ++mt) {
        const float* arow = Tp + (size_t)(mt * 16 + nn) * DIMN + 2 * kh;
        v8f acc = {};
#pragma unroll 4
        for (int k = 0; k < DIMN; k += 4) {
            v2f a = *(const v2f*)(arow + k);
            v2f b = *(const v2f*)(brow + k);
            __builtin_prefetch(brow + k + 64, 0, 0);
            acc = __builtin_amdgcn_wmma_f32_16x16x4_f32(
                false, a, false, b, (short)0, acc, false, false);
        }
        float buf[8];
        for (int r = 0; r < 8; ++r) {
            int   w    = mt * 16 + (kh << 3) + r;
            float sc   = ws[OFF_SC + w];
            float cc   = ws[OFF_C + w];
            float maxn = ws[OFF_MAXN + w];
            float u2   = ws[OFF_TN2 + w];
            float beta = expmap_scale(nx, sc, maxn);
            float y2   = beta * beta * qn2;
            float dot  = beta * acc[r];
            buf[r] = -mob_dist(dot, u2, y2, cc, sc) / TEMPF;
        }
        // w = mt*16 + kh*8 + r  -> consecutive outputs at out[q*32 + ...]
        float* dst = out + (size_t)q * WAYN + mt * 16 + (kh << 3);
        *(float4*)(dst + 0) = make_float4(buf[0], buf[1], buf[2], buf[3]);
        *(float4*)(dst + 4) = make_float4(buf[4], buf[5], buf[6], buf[7]);
    }
}

// ---------------- launcher ----------------
extern "C" void kernel_launch(void* const* d_in, const int* in_sizes, int n_in,
                              void* d_out, int out_size, void* d_ws, size_t ws_size,
                              hipStream_t stream) {
    const float* shot  = (const float*)d_in[0];
    const float* query = (const float*)d_in[1];
    const float* W1  = (const float*)d_in[2];
    const float* b1  = (const float*)d_in[3];
    const float* W2  = (const float*)d_in[4];
    const float* b2  = (const float*)d_in[5];
    const float* W3  = (const float*)d_in[6];
    const float* b3  = (const float*)d_in[7];
    const float* Wr1 = (const float*)d_in[8];
    const float* br1 = (const float*)d_in[9];
    const float* Wr2 = (const float*)d_in[10];
    const float* br2 = (const float*)d_in[11];
    float* ws  = (float*)d_ws;
    float* out = (float*)d_out;

    k_prep    <<<(DIMN + 127) / 128, 128, 0, stream>>>(shot, query, ws);
    k_qn2     <<<NQ / 8, 256, 0, stream>>>(query, ws);
    k_ctrl    <<<1, 256, 0, stream>>>(W1, b1, W2, b2, W3, b3, ws);
    k_gemm_dis<<<NQ / 128, 256, 0, stream>>>(query, ws);
    k_rowstats<<<WAYN, 256, 0, stream>>>(ws);
    k_relnet  <<<1, 256, 0, stream>>>(query, Wr1, br1, Wr2, br2, ws);
    k_gemm_out<<<NQ / 128, 256, 0, stream>>>(query, ws, out);
}